// EdgeLengthLoss_62294205661547
// MI455X (gfx1250) — compile-verified
//
#include <hip/hip_runtime.h>

typedef __attribute__((ext_vector_type(2))) float v2f;
typedef __attribute__((ext_vector_type(8))) float v8f;

// Exact f32 wave(32)-wide sum using two chained V_WMMA_F32_16X16X4_F32.
// A-matrix 16x4 layout (ISA 7.12.2): lane L<16 supplies (M=L,K=0),(M=L,K=1);
// lane L>=16 supplies (M=L-16,K=2),(M=L-16,K=3).  With A = {local, 0} and
// B = ones(4x16), D[m,n] = local_m + local_{m+16}.
// D layout: VGPR j holds row M=j (lanes 0-15) / M=j+8 (lanes 16-31), so
// summing a lane's 8 D VGPRs yields the half-wave totals (lanes 0-15 get
// rows 0-7 => lanes {0..7,16..23}; lanes 16-31 get the complement).
// A second identical WMMA adds the two halves, leaving the full wave sum
// replicated in every lane of every D VGPR.
// REQUIRES: EXEC all ones (no divergence at the call site).
__device__ __forceinline__ float wave_reduce_wmma(float local) {
    v2f a;    a[0] = local; a[1] = 0.0f;
    v2f ones; ones[0] = 1.0f; ones[1] = 1.0f;
    v8f c = {};
    v8f d = __builtin_amdgcn_wmma_f32_16x16x4_f32(
        /*neg_a=*/false, a, /*neg_b=*/false, ones,
        /*c_mod=*/(short)0, c, /*reuse_a=*/false, /*reuse_b=*/false);
    float half = d[0] + d[1] + d[2] + d[3] + d[4] + d[5] + d[6] + d[7];
    v2f a2; a2[0] = half; a2[1] = 0.0f;
    v8f d2 = __builtin_amdgcn_wmma_f32_16x16x4_f32(
        false, a2, false, ones, (short)0, c, false, false);
    return d2[0];
}

__device__ __forceinline__ float edge_len(const float* __restrict__ p,
                                          const float* __restrict__ q) {
    float dx = p[0] - q[0];
    float dy = p[1] - q[1];
    float dz = p[2] - q[2];
    return sqrtf(dx * dx + dy * dy + dz * dz);
}

__global__ void __launch_bounds__(256)
edge_length_loss_kernel(const float* __restrict__ coord_out,
                        const float* __restrict__ coord_gt,
                        const float* __restrict__ weight,
                        const int*   __restrict__ face,
                        float* __restrict__ out,
                        int F, int V, float inv_den) {
    const int f = blockIdx.x * blockDim.x + threadIdx.x;
    const int b = blockIdx.y;

    float local = 0.0f;
    if (f < F) {   // no early return: EXEC must be all-ones for the WMMA below
        const int i0 = face[3 * f + 0];
        const int i1 = face[3 * f + 1];
        const int i2 = face[3 * f + 2];

        float w0 = weight[i0];
        float w1 = weight[i1];
        float w2 = weight[i2];
        float w01 = w0 * w1; if (w01 == 1.0f) w01 = 0.0f;
        float w02 = w0 * w2; if (w02 == 1.0f) w02 = 0.0f;
        float w12 = w1 * w2; if (w12 == 1.0f) w12 = 0.0f;

        const size_t base = (size_t)b * (size_t)V * 3u;
        const float* o0 = coord_out + base + (size_t)i0 * 3u;
        const float* o1 = coord_out + base + (size_t)i1 * 3u;
        const float* o2 = coord_out + base + (size_t)i2 * 3u;
        const float* g0 = coord_gt  + base + (size_t)i0 * 3u;
        const float* g1 = coord_gt  + base + (size_t)i1 * 3u;
        const float* g2 = coord_gt  + base + (size_t)i2 * 3u;

        float d01 = fabsf(edge_len(o0, o1) - edge_len(g0, g1)) * w01;
        float d02 = fabsf(edge_len(o0, o2) - edge_len(g0, g2)) * w02;
        float d12 = fabsf(edge_len(o1, o2) - edge_len(g1, g2)) * w12;
        local = d01 + d02 + d12;
    }

    // Wave-level exact reduction on the matrix pipe.
    const float wave_sum = wave_reduce_wmma(local);

    // Cross-wave combine through LDS, one atomic per block.
    __shared__ float smem[8];
    const int wave = threadIdx.x >> 5;
    const int lane = threadIdx.x & 31;
    if (lane == 0) smem[wave] = wave_sum;
    __syncthreads();
    if (threadIdx.x == 0) {
        float s = 0.0f;
        const int nwaves = blockDim.x >> 5;
        for (int i = 0; i < nwaves; ++i) s += smem[i];
        atomicAdd(out, s * inv_den);
    }
}

__global__ void zero_scalar_kernel(float* __restrict__ out) {
    if (threadIdx.x == 0 && blockIdx.x == 0) out[0] = 0.0f;
}

extern "C" void kernel_launch(void* const* d_in, const int* in_sizes, int n_in,
                              void* d_out, int out_size, void* d_ws, size_t ws_size,
                              hipStream_t stream) {
    (void)n_in; (void)out_size; (void)d_ws; (void)ws_size;

    const float* coord_out = (const float*)d_in[0];
    const float* coord_gt  = (const float*)d_in[1];
    const float* weight    = (const float*)d_in[2];
    const int*   face      = (const int*)d_in[3];
    float* out = (float*)d_out;

    const int V = in_sizes[2];
    const int F = in_sizes[3] / 3;
    const int B = in_sizes[0] / (V * 3);
    const float inv_den = (float)(1.0 / ((double)B * 3.0 * (double)F));

    zero_scalar_kernel<<<1, 32, 0, stream>>>(out);

    dim3 block(256, 1, 1);
    dim3 grid((unsigned)((F + 255) / 256), (unsigned)B, 1);
    edge_length_loss_kernel<<<grid, block, 0, stream>>>(
        coord_out, coord_gt, weight, face, out, F, V, inv_den);
}